// ArcMarginProduct_17162689315310
// MI455X (gfx1250) — compile-verified
//
#include <hip/hip_runtime.h>
#include <math.h>

typedef float v2f __attribute__((ext_vector_type(2)));
typedef float v8f __attribute__((ext_vector_type(8)));

#define IN_F   512
#define OUT_F  50000
#define BATCH  4096
#define SCALE_ 30.0f
#define COS_M_ 0.87758256189037276f
#define SIN_M_ 0.47942553860420301f
#define TH_C   (-0.87758256189037276f)
#define MM_C   0.23971276930210156f

#define MT 64
#define NT 80
#define KC 32
#define KP 36   // padded LDS row stride (floats): 16B-aligned rows, spreads banks

// ---------------- inverse L2 row norms ----------------
__global__ void rownorm_kernel(const float* __restrict__ x,
                               float* __restrict__ inv, int ncols) {
  __shared__ float red[256];
  const int row = blockIdx.x;
  const float* p = x + (size_t)row * ncols;
  float s = 0.0f;
  for (int i = threadIdx.x; i < ncols; i += blockDim.x) {
    float v = p[i];
    s += v * v;
  }
  red[threadIdx.x] = s;
  __syncthreads();
  for (int off = 128; off > 0; off >>= 1) {
    if (threadIdx.x < off) red[threadIdx.x] += red[threadIdx.x + off];
    __syncthreads();
  }
  if (threadIdx.x == 0) {
    float n = sqrtf(red[0]);
    inv[row] = 1.0f / fmaxf(n, 1e-12f);
  }
}

// ---------------- WMMA GEMM + ArcFace epilogue ----------------
// C[64x80] per block of 4 waves; wave w owns rows [w*16, w*16+16) x 80 cols
// = 5 tiles of 16x16. K=512 swept in chunks of 32 with double-buffered LDS:
// next chunk's global loads are issued before the barrier and overlap the
// v_wmma stream; ds_store into the other buffer after compute.
__global__ __launch_bounds__(128) void arc_gemm_kernel(
    const float* __restrict__ A,     // [BATCH][IN_F]  raw input
    const float* __restrict__ W,     // [OUT_F][IN_F]  raw weight
    const float* __restrict__ invx,  // [BATCH]
    const float* __restrict__ invw,  // [OUT_F]
    const int*   __restrict__ label, // [BATCH]
    float* __restrict__ out)         // [BATCH][OUT_F]
{
  __shared__ float As[2][MT][KP];
  __shared__ float Bs[2][NT][KP];

  const int tid  = threadIdx.x;
  const int wave = tid >> 5;
  const int lane = tid & 31;
  const int ln   = lane & 15;   // N (B,C,D) / M (A) index within half-wave
  const int g    = lane >> 4;   // half-wave group

  const int m0 = blockIdx.y * MT;
  const int n0 = blockIdx.x * NT;

  v8f acc[5];
  const v8f vzero = {0.f,0.f,0.f,0.f,0.f,0.f,0.f,0.f};
#pragma unroll
  for (int t = 0; t < 5; ++t) acc[t] = vzero;

  float4 areg[4];
  float4 breg[5];

  // Prologue: fetch K-chunk 0 and stage into buffer 0.
#pragma unroll
  for (int i = 0; i < 4; ++i) {
    int idx = tid + i * 128;
    int r = idx >> 3, kq = idx & 7;
    areg[i] = *(const float4*)(A + (size_t)(m0 + r) * IN_F + kq * 4);
  }
#pragma unroll
  for (int i = 0; i < 5; ++i) {
    int idx = tid + i * 128;
    int c = idx >> 3, kq = idx & 7;
    breg[i] = *(const float4*)(W + (size_t)(n0 + c) * IN_F + kq * 4);
  }
#pragma unroll
  for (int i = 0; i < 4; ++i) {
    int idx = tid + i * 128;
    int r = idx >> 3, kq = idx & 7;
    *(float4*)&As[0][r][kq * 4] = areg[i];
  }
#pragma unroll
  for (int i = 0; i < 5; ++i) {
    int idx = tid + i * 128;
    int c = idx >> 3, kq = idx & 7;
    *(float4*)&Bs[0][c][kq * 4] = breg[i];
  }

  int p = 0;
  for (int kc = 0; kc < IN_F; kc += KC) {
    const bool more = (kc + KC) < IN_F;
    if (more) {
      // Issue next chunk's global loads; they stay in flight over the WMMAs.
#pragma unroll
      for (int i = 0; i < 4; ++i) {
        int idx = tid + i * 128;
        int r = idx >> 3, kq = idx & 7;
        areg[i] = *(const float4*)(A + (size_t)(m0 + r) * IN_F + kc + KC + kq * 4);
      }
#pragma unroll
      for (int i = 0; i < 5; ++i) {
        int idx = tid + i * 128;
        int c = idx >> 3, kq = idx & 7;
        breg[i] = *(const float4*)(W + (size_t)(n0 + c) * IN_F + kc + KC + kq * 4);
      }
    }
    __syncthreads();   // buffer p ready for all waves; buffer 1-p free to fill

#pragma unroll
    for (int kk = 0; kk < KC; kk += 4) {
      // A fragment 16x4: lane ln = M, VGPR pair holds K = kk+2g, kk+2g+1
      v2f a;
      a.x = As[p][wave * 16 + ln][kk + 2 * g];
      a.y = As[p][wave * 16 + ln][kk + 2 * g + 1];
#pragma unroll
      for (int t = 0; t < 5; ++t) {
        // B fragment 4x16: lane ln = N, VGPR v holds K = kk + v + 2g
        v2f b;
        b.x = Bs[p][t * 16 + ln][kk + 2 * g];
        b.y = Bs[p][t * 16 + ln][kk + 2 * g + 1];
        acc[t] = __builtin_amdgcn_wmma_f32_16x16x4_f32(
            false, a, false, b, (short)0, acc[t], false, false);
      }
    }

    if (more) {
#pragma unroll
      for (int i = 0; i < 4; ++i) {
        int idx = tid + i * 128;
        int r = idx >> 3, kq = idx & 7;
        *(float4*)&As[1 - p][r][kq * 4] = areg[i];
      }
#pragma unroll
      for (int i = 0; i < 5; ++i) {
        int idx = tid + i * 128;
        int c = idx >> 3, kq = idx & 7;
        *(float4*)&Bs[1 - p][c][kq * 4] = breg[i];
      }
    }
    p ^= 1;
  }

  // Epilogue: cosine = acc * invx[row] * invw[col]; ArcFace margin; scale.
  // Non-temporal stores: the 819MB logit stream must not evict the
  // L2-resident weight matrix (102MB of 192MB L2).
  float ivx[8];
  int   lab[8];
#pragma unroll
  for (int v = 0; v < 8; ++v) {
    int row = m0 + wave * 16 + v + 8 * g;   // C/D layout: M = VGPR + 8*group
    ivx[v] = invx[row];
    lab[v] = label[row];
  }
#pragma unroll
  for (int t = 0; t < 5; ++t) {
    int col = n0 + t * 16 + ln;
    float iw = invw[col];
#pragma unroll
    for (int v = 0; v < 8; ++v) {
      int row = m0 + wave * 16 + v + 8 * g;
      float c = acc[t][v] * ivx[v] * iw;
      float sine = sqrtf(1.0f - c * c);          // matches reference (NaN if |c|>1)
      float phi  = c * COS_M_ - sine * SIN_M_;
      phi = (c > TH_C) ? phi : (c - MM_C);
      float o = (lab[v] == col) ? phi : c;
      __builtin_nontemporal_store(o * SCALE_, &out[(size_t)row * OUT_F + col]);
    }
  }
}

// ---------------- per-row online-softmax LSE -> row loss ----------------
__global__ void row_lse_kernel(const float* __restrict__ out,
                               const int* __restrict__ label,
                               float* __restrict__ row_loss) {
  __shared__ float mred[256];
  __shared__ float sred[256];
  const int row = blockIdx.x;
  const float* p = out + (size_t)row * OUT_F;
  float m = -3.0e38f, s = 0.0f;
  for (int i = threadIdx.x; i < OUT_F; i += blockDim.x) {
    float x = __builtin_nontemporal_load(p + i);  // streaming re-read, keep L2 clean
    if (x > m) { s *= __expf(m - x); m = x; }
    s += __expf(x - m);
  }
  mred[threadIdx.x] = m;
  sred[threadIdx.x] = s;
  __syncthreads();
  for (int off = 128; off > 0; off >>= 1) {
    if (threadIdx.x < off) {
      float m1 = mred[threadIdx.x],       s1 = sred[threadIdx.x];
      float m2 = mred[threadIdx.x + off], s2 = sred[threadIdx.x + off];
      float M = fmaxf(m1, m2);
      sred[threadIdx.x] = s1 * __expf(m1 - M) + s2 * __expf(m2 - M);
      mred[threadIdx.x] = M;
    }
    __syncthreads();
  }
  if (threadIdx.x == 0) {
    float lse = mred[0] + __logf(sred[0]);
    row_loss[row] = lse - p[label[row]];
  }
}

__global__ void loss_reduce_kernel(const float* __restrict__ row_loss,
                                   float* __restrict__ loss_out) {
  __shared__ float red[256];
  float s = 0.0f;
  for (int i = threadIdx.x; i < BATCH; i += 256) s += row_loss[i];
  red[threadIdx.x] = s;
  __syncthreads();
  for (int off = 128; off > 0; off >>= 1) {
    if (threadIdx.x < off) red[threadIdx.x] += red[threadIdx.x + off];
    __syncthreads();
  }
  if (threadIdx.x == 0) *loss_out = red[0] * (1.0f / (float)BATCH);
}

extern "C" void kernel_launch(void* const* d_in, const int* in_sizes, int n_in,
                              void* d_out, int out_size, void* d_ws, size_t ws_size,
                              hipStream_t stream) {
  const float* input  = (const float*)d_in[0];
  const int*   label  = (const int*)d_in[1];
  const float* weight = (const float*)d_in[2];
  float* out = (float*)d_out;

  float* ws    = (float*)d_ws;
  float* invx  = ws;                    // BATCH floats
  float* invw  = ws + BATCH;            // OUT_F floats
  float* rloss = ws + BATCH + OUT_F;    // BATCH floats

  rownorm_kernel<<<BATCH, 256, 0, stream>>>(input, invx, IN_F);
  rownorm_kernel<<<OUT_F, 256, 0, stream>>>(weight, invw, IN_F);

  dim3 grid(OUT_F / NT, BATCH / MT);    // 625 x 64 blocks
  arc_gemm_kernel<<<grid, 128, 0, stream>>>(input, weight, invx, invw, label, out);

  row_lse_kernel<<<BATCH, 256, 0, stream>>>(out, label, rloss);
  loss_reduce_kernel<<<1, 256, 0, stream>>>(rloss, out + (size_t)BATCH * OUT_F);
}